// Unit_test_sim_30485677868016
// MI455X (gfx1250) — compile-verified
//
#include <hip/hip_runtime.h>
#include <stdint.h>

// Problem constants (match reference: B=2048, T=512, NV=16, DT=0.01)
#define B_DIM   2048
#define T_DIM   512
#define ROW     48                    // NV*3 floats per (b,t) row
#define DT_F    0.01f
#define INV_DT  100.0f
#define TT      64                    // timesteps per LDS tile
#define NTILES  (T_DIM / TT)          // 8
#define TPB     192                   // 6 wave32 waves
#define J4S     12                    // float4 columns per 48-float row
#define TGROUPS (TPB / J4S)           // 16 t-groups
#define RPT     (TT / TGROUPS)        // 4 rows per thread per tile
#define TILE_F  (TT * ROW)            // 3072 floats per stream per tile
#define Q_STREAM (TILE_F / 4)         // 768 float4 per stream per tile
#define LPT     (Q_STREAM / TPB)      // 4 async b128 copies per thread per stream

// Exact parameter types of the async-copy builtin on this toolchain:
// (v4i addrspace(1)*, v4i addrspace(3)*, imm int, imm int)
typedef int v4i __attribute__((vector_size(16)));
typedef __attribute__((address_space(1))) v4i* gptr_b128;
typedef __attribute__((address_space(3))) v4i* lptr_b128;

// ---- CDNA5 async global->LDS copy (ASYNCcnt path, CDNA5 ISA) ----
__device__ __forceinline__ void async_cp16(const float* gbase, unsigned byteoff,
                                           const float* ldsdst) {
#if __has_builtin(__builtin_amdgcn_global_load_async_to_lds_b128)
  __builtin_amdgcn_global_load_async_to_lds_b128(
      (gptr_b128)((unsigned long long)(uintptr_t)gbase + (unsigned long long)byteoff),
      (lptr_b128)(unsigned int)(uintptr_t)ldsdst,   // low 32 bits of generic LDS ptr = LDS offset
      0, 0);
#else
  unsigned int l = (unsigned int)(uintptr_t)ldsdst;
  asm volatile("global_load_async_to_lds_b128 %0, %1, %2"
               :: "v"(l), "v"(byteoff), "s"(gbase)
               : "memory");
#endif
}

#if __has_builtin(__builtin_amdgcn_s_wait_asynccnt)
#define WAIT_ASYNC(n) __builtin_amdgcn_s_wait_asynccnt(n)
#else
#define WAIT_ASYNC(n) asm volatile("s_wait_asynccnt %0" :: "i"(n) : "memory")
#endif

// One block per batch b. Streams positions_traj[b] and target_traj[b] through
// double-buffered LDS via async copies; closed-form Verlet state in registers.
__global__ __launch_bounds__(TPB) void verlet_loss_kernel(
    const float* __restrict__ pos_traj,   // (B,T,NV,3)
    const float* __restrict__ prev_traj,  // (B,T,NV,3) -- only t=0 row used
    const float* __restrict__ tgt_traj,   // (B,T,NV,3)
    const float* __restrict__ gravity,    // (3,)
    float* __restrict__ partials)         // (B,2) unscaled [sum_pos_sq, sum_vel_sq]
{
  __shared__ float sPos[2][TILE_F];
  __shared__ float sTgt[2][TILE_F];
  __shared__ float sRedP[TPB];
  __shared__ float sRedV[TPB];

  const int tid  = threadIdx.x;
  const int b    = blockIdx.x;
  const int j4   = tid % J4S;           // which float4 column of the 48-float row
  const int tgrp = tid / J4S;           // which t-row group

  const size_t bbase = (size_t)b * T_DIM * ROW;
  const float* bpos = pos_traj + bbase;
  const float* btgt = tgt_traj + bbase;

  // Initial state (t=0 row): A = pos0, Vd = pos0 - prev0 (= v0*dt). Mass cancels.
  const float4 A = *(const float4*)(bpos + j4 * 4);
  const float4 P = *(const float4*)(prev_traj + bbase + j4 * 4);
  const float4 Vd = make_float4(A.x - P.x, A.y - P.y, A.z - P.z, A.w - P.w);

  // Gravity component per element of this thread's float4 (axis = (4*j4+e) % 3).
  const float g0 = gravity[0], g1 = gravity[1], g2 = gravity[2];
  const int m0 = (j4 * 4) % 3;
  const int m1 = (m0 + 1) % 3;
  const int m2 = (m1 + 1) % 3;
  const float ge0 = (m0 == 0) ? g0 : ((m0 == 1) ? g1 : g2);
  const float ge1 = (m1 == 0) ? g0 : ((m1 == 1) ? g1 : g2);
  const float ge2 = (m2 == 0) ? g0 : ((m2 == 1) ? g1 : g2);
  const float ge3 = ge0;                 // (4*j4+3) % 3 == (4*j4) % 3

  auto issue = [&](int tile, int buf) {
    const unsigned tile_off = (unsigned)tile * (TILE_F * 4u);  // bytes
#pragma unroll
    for (int k = 0; k < LPT; ++k) {
      const int q = tid + k * TPB;                  // float4 index within tile
      async_cp16(bpos, tile_off + (unsigned)q * 16u, &sPos[buf][q * 4]);
      async_cp16(btgt, tile_off + (unsigned)q * 16u, &sTgt[buf][q * 4]);
    }
  };

  float sp = 0.0f, sv = 0.0f;
  issue(0, 0);

  for (int i = 0; i < NTILES; ++i) {
    if (i + 1 < NTILES) { issue(i + 1, (i + 1) & 1); WAIT_ASYNC(2 * LPT); }
    else                { WAIT_ASYNC(0); }
    __syncthreads();                      // tile i fully resident for all threads
    const int buf = i & 1;
#pragma unroll
    for (int r = 0; r < RPT; ++r) {
      const int trow = tgrp + r * TGROUPS;
      const int t = i * TT + trow;
      const float s1 = (float)(t + 1);
      const float c1 = 2.0f * s1 - 1.0f;                       // Vd coefficient
      const float c2 = 0.5f * DT_F * DT_F * s1 * (s1 + 1.0f);  // g coefficient (pos)
      const float c3 = DT_F * s1;                              // g coefficient (vel)
      const float4 T4 = *(const float4*)&sTgt[buf][trow * ROW + j4 * 4];
      const float4 P4 = *(const float4*)&sPos[buf][trow * ROW + j4 * 4];
#define ACC_ELEM(Ae, Ve, Ge, Te, Pe)                                          \
      { const float pn = Ae + Ve * c1 + Ge * c2;                              \
        const float ep = pn - (Te);                                           \
        const float ev = (Ve) * (2.0f * INV_DT) + Ge * c3 - ((Te) - (Pe)) * INV_DT; \
        sp = fmaf(ep, ep, sp); sv = fmaf(ev, ev, sv); }
      ACC_ELEM(A.x, Vd.x, ge0, T4.x, P4.x)
      ACC_ELEM(A.y, Vd.y, ge1, T4.y, P4.y)
      ACC_ELEM(A.z, Vd.z, ge2, T4.z, P4.z)
      ACC_ELEM(A.w, Vd.w, ge3, T4.w, P4.w)
#undef ACC_ELEM
    }
    __syncthreads();                      // tile i consumed before its buffer is reused
  }

  // Block reduction (192 threads)
  sRedP[tid] = sp; sRedV[tid] = sv;
  __syncthreads();
  for (int s = 96; s >= 6; s >>= 1) {
    if (tid < s) { sRedP[tid] += sRedP[tid + s]; sRedV[tid] += sRedV[tid + s]; }
    __syncthreads();
  }
  if (tid == 0) {
    float tp = 0.0f, tv = 0.0f;
#pragma unroll
    for (int k = 0; k < 6; ++k) { tp += sRedP[k]; tv += sRedV[k]; }
    partials[2 * b + 0] = tp;
    partials[2 * b + 1] = tv;
  }
}

// Deterministic final reduce: 2048 partial pairs -> two scalars.
__global__ __launch_bounds__(256) void reduce_partials(
    const float* __restrict__ partials, float* __restrict__ out)
{
  __shared__ float rp[256], rv[256];
  const int tid = threadIdx.x;
  float tp = 0.0f, tv = 0.0f;
  for (int i = tid; i < B_DIM; i += 256) {
    tp += partials[2 * i + 0];
    tv += partials[2 * i + 1];
  }
  rp[tid] = tp; rv[tid] = tv;
  __syncthreads();
  for (int s = 128; s > 0; s >>= 1) {
    if (tid < s) { rp[tid] += rp[tid + s]; rv[tid] += rv[tid + s]; }
    __syncthreads();
  }
  if (tid == 0) {
    const float inv = 1.0f / (float)(B_DIM * 16 * 3);   // mean over (B,NV,3) per step
    out[0] = rp[0] * inv;                                // sum_t step_loss_pos
    out[1] = (rp[0] + rv[0]) * inv;                      // sum_t (loss_pos + loss_vel)
  }
}

extern "C" void kernel_launch(void* const* d_in, const int* in_sizes, int n_in,
                              void* d_out, int out_size, void* d_ws, size_t ws_size,
                              hipStream_t stream) {
  const float* pos_traj  = (const float*)d_in[0];
  const float* prev_traj = (const float*)d_in[1];
  const float* tgt_traj  = (const float*)d_in[2];
  // d_in[3] mass_diagonal: mathematically cancels (acc = m*g/m = g)
  const float* gravity   = (const float*)d_in[4];
  // d_in[5] time_horizon: fixed at T_DIM

  float* partials = (float*)d_ws;           // needs B_DIM*2*4 = 16 KB of scratch
  float* out      = (float*)d_out;

  verlet_loss_kernel<<<B_DIM, TPB, 0, stream>>>(pos_traj, prev_traj, tgt_traj,
                                                gravity, partials);
  reduce_partials<<<1, 256, 0, stream>>>(partials, out);
}